// GraphDenoiser_17025250361519
// MI455X (gfx1250) — compile-verified
//
#include <hip/hip_runtime.h>
#include <math.h>

#define NND 768   // N nodes
#define FD  256   // input feature dim
#define HD  128   // hidden dim

typedef __attribute__((ext_vector_type(2))) float v2f;
typedef __attribute__((ext_vector_type(8))) float v8f;

#if __has_builtin(__builtin_amdgcn_wmma_f32_16x16x4_f32)
#define HAVE_WMMA_F32 1
#else
#define HAVE_WMMA_F32 0
#endif

// ---------------------------------------------------------------------------
// 1) d_inv_sqrt over rows of (adj + I)
// ---------------------------------------------------------------------------
__global__ __launch_bounds__(256) void rowsum_kernel(const float* __restrict__ adj,
                                                     float* __restrict__ dinv) {
  __shared__ float red[256];
  const int i = blockIdx.x;
  float s = 0.0f;
  for (int j = threadIdx.x; j < NND; j += 256) s += adj[i * NND + j];
  red[threadIdx.x] = s;
  __syncthreads();
  for (int off = 128; off > 0; off >>= 1) {
    if (threadIdx.x < off) red[threadIdx.x] += red[threadIdx.x + off];
    __syncthreads();
  }
  if (threadIdx.x == 0) {
    const float d = red[0] + 1.0f;  // + self loop
    dinv[i] = (d > 0.0f) ? rsqrtf(d) : 0.0f;
  }
}

// ---------------------------------------------------------------------------
// 2) a_norm = dinv[i] * (adj + I) * dinv[j]
// ---------------------------------------------------------------------------
__global__ __launch_bounds__(256) void norm_adj_kernel(const float* __restrict__ adj,
                                                       const float* __restrict__ dinv,
                                                       float* __restrict__ anorm) {
  const int idx = blockIdx.x * 256 + threadIdx.x;
  const int i = idx / NND, j = idx % NND;
  const float a = adj[idx] + ((i == j) ? 1.0f : 0.0f);
  anorm[idx] = dinv[i] * a * dinv[j];
}

// ---------------------------------------------------------------------------
// 3) f32 GEMM via V_WMMA_F32_16X16X4_F32. One wave = one 16x16 C tile.
//    A: MxK row-major, B: KxN row-major, C: MxN row-major. K % 4 == 0.
//    Tile choice is wave-uniform -> EXEC all-1s across WMMA (ISA §7.12 req).
// ---------------------------------------------------------------------------
__global__ __launch_bounds__(256) void wmma_gemm_f32(const float* __restrict__ A,
                                                     const float* __restrict__ B,
                                                     float* __restrict__ C,
                                                     int M, int N, int K, int relu) {
  const int lane   = threadIdx.x & 31;
  const int wave   = threadIdx.x >> 5;
  const int tilesN = N >> 4;
  const int tile   = blockIdx.x * (blockDim.x >> 5) + wave;
  if (tile >= (M >> 4) * tilesN) return;  // wave-uniform guard
  const int tm = (tile / tilesN) << 4;
  const int tn = (tile % tilesN) << 4;
  const int lhalf = lane >> 4;   // 0: K lanes 0..1 ; 1: K lanes 2..3
  const int l16   = lane & 15;

  v8f acc = {};
  // A fragment: lane row = tm + l16; VGPR0 = K + 2*lhalf, VGPR1 = K + 2*lhalf + 1
  const float* arow = A + (size_t)(tm + l16) * K + (lhalf << 1);
  const float* bcol = B + tn + l16;

  for (int k = 0; k < K; k += 4) {
    v2f a;
    a.x = arow[k];
    a.y = arow[k + 1];
    v2f b;
    b.x = bcol[(size_t)(k + (lhalf << 1) + 0) * N];
    b.y = bcol[(size_t)(k + (lhalf << 1) + 1) * N];
#if HAVE_WMMA_F32
    acc = __builtin_amdgcn_wmma_f32_16x16x4_f32(false, a, false, b,
                                                (short)0, acc, false, false);
#else
    // Semantically identical fallback (matches WMMA C/D lane layout)
    #pragma unroll
    for (int v = 0; v < 8; ++v) {
      const int row = tm + v + (lhalf << 3);
      float s = acc[v];
      for (int kk = k; kk < k + 4; ++kk)
        s += A[(size_t)row * K + kk] * B[(size_t)kk * N + tn + l16];
      acc[v] = s;
    }
#endif
  }

  // C/D layout: VGPR v -> row tm + v + 8*lhalf, column tn + l16
  float* crow = C + tn + l16;
  #pragma unroll
  for (int v = 0; v < 8; ++v) {
    float val = acc[v];
    if (relu) val = fmaxf(val, 0.0f);
    crow[(size_t)(tm + v + (lhalf << 3)) * N] = val;
  }
}

// ---------------------------------------------------------------------------
// 4) c = time_emb[t] @ we1[2H:] + be1   (128x128 matvec, one block)
// ---------------------------------------------------------------------------
__global__ __launch_bounds__(128) void cvec_kernel(const float* __restrict__ time_emb,
                                                   const float* __restrict__ we1,
                                                   const float* __restrict__ be1,
                                                   const int* __restrict__ t_ptr,
                                                   float* __restrict__ cvec) {
  const int k = threadIdx.x;
  const int t = t_ptr[0];
  const float* te = time_emb + (size_t)t * HD;
  float acc = 0.0f;
  for (int m = 0; m < HD; ++m)
    acc += te[m] * we1[(size_t)(2 * HD + m) * HD + k];
  cvec[k] = acc + be1[k];
}

// ---------------------------------------------------------------------------
// 5) S[i][j] = sigmoid(be2 + sum_k relu(p[i,k]+q[j,k]+c[k]) * we2[k])
//    16x16 (i,j) tile per block; p/q tiles staged in LDS (padded vs conflicts)
// ---------------------------------------------------------------------------
__global__ __launch_bounds__(256) void edge_kernel(const float* __restrict__ p,
                                                   const float* __restrict__ q,
                                                   const float* __restrict__ cvec,
                                                   const float* __restrict__ we2,
                                                   const float* __restrict__ be2,
                                                   float* __restrict__ S) {
  __shared__ float ps[16][HD + 1];
  __shared__ float qs[16][HD + 1];
  __shared__ float cs[HD];
  __shared__ float w2s[HD];
  const int i0 = blockIdx.y << 4;
  const int j0 = blockIdx.x << 4;
  const int tid = threadIdx.x;

  for (int idx = tid; idx < 16 * HD; idx += 256) {
    const int r = idx >> 7, k = idx & (HD - 1);
    ps[r][k] = p[(size_t)(i0 + r) * HD + k];
    qs[r][k] = q[(size_t)(j0 + r) * HD + k];
  }
  if (tid < HD) { cs[tid] = cvec[tid]; w2s[tid] = we2[tid]; }
  __syncthreads();

  const int li = tid >> 4;
  const int lj = tid & 15;
  float acc = be2[0];
  #pragma unroll 4
  for (int k = 0; k < HD; ++k) {
    const float z = ps[li][k] + qs[lj][k] + cs[k];
    acc = fmaf(fmaxf(z, 0.0f), w2s[k], acc);
  }
  S[(size_t)(i0 + li) * NND + (j0 + lj)] = 1.0f / (1.0f + expf(-acc));
}

// ---------------------------------------------------------------------------
// 6) out = (S + S^T) / 2
// ---------------------------------------------------------------------------
__global__ __launch_bounds__(256) void symmetrize_kernel(const float* __restrict__ S,
                                                         float* __restrict__ out) {
  const int idx = blockIdx.x * 256 + threadIdx.x;
  const int i = idx / NND, j = idx % NND;
  out[idx] = 0.5f * (S[idx] + S[(size_t)j * NND + i]);
}

// ---------------------------------------------------------------------------
extern "C" void kernel_launch(void* const* d_in, const int* in_sizes, int n_in,
                              void* d_out, int out_size, void* d_ws, size_t ws_size,
                              hipStream_t stream) {
  const float* x    = (const float*)d_in[0];   // (768, 256)
  const float* adj  = (const float*)d_in[1];   // (768, 768)
  const float* w1   = (const float*)d_in[2];   // (256, 128)
  const float* w2   = (const float*)d_in[3];   // (128, 128)
  const float* temb = (const float*)d_in[4];   // (100, 128)
  const float* we1  = (const float*)d_in[5];   // (384, 128)
  const float* be1  = (const float*)d_in[6];   // (128,)
  const float* we2  = (const float*)d_in[7];   // (128, 1)
  const float* be2  = (const float*)d_in[8];   // (1,)
  const int*   t    = (const int*)d_in[9];     // scalar

  float* ws    = (float*)d_ws;
  float* dinv  = ws;                       // 768
  float* cvec  = ws + 1024;                // 128
  float* anorm = ws + 2048;                // 768*768
  float* S     = anorm + NND * NND;        // 768*768
  float* buf1  = S + NND * NND;            // 768*128 : t1 -> t2 -> p
  float* buf2  = buf1 + NND * HD;          // 768*128 : h1 -> h2
  float* buf3  = buf2 + NND * HD;          // 768*128 : q

  rowsum_kernel<<<NND, 256, 0, stream>>>(adj, dinv);
  norm_adj_kernel<<<(NND * NND) / 256, 256, 0, stream>>>(adj, dinv, anorm);

  // All GEMMs: N=128 -> 48x8 = 384 tiles -> 48 blocks of 8 waves
  #define GEMM(A, B, C, M, Nn, K, relu) \
    wmma_gemm_f32<<<((M / 16) * (Nn / 16)) / 8, 256, 0, stream>>>(A, B, C, M, Nn, K, relu)

  GEMM(x,     w1,            buf1, NND, HD, FD,  0);  // t1 = x @ w1
  GEMM(anorm, buf1,          buf2, NND, HD, NND, 1);  // h1 = relu(a_norm @ t1)
  GEMM(buf2,  w2,            buf1, NND, HD, HD,  0);  // t2 = h1 @ w2
  GEMM(anorm, buf1,          buf2, NND, HD, NND, 1);  // h2 = relu(a_norm @ t2)
  GEMM(buf2,  we1,           buf1, NND, HD, HD,  0);  // p  = h2 @ we1[:H]
  GEMM(buf2,  we1 + HD * HD, buf3, NND, HD, HD,  0);  // q  = h2 @ we1[H:2H]
  #undef GEMM

  cvec_kernel<<<1, HD, 0, stream>>>(temb, we1, be1, t, cvec);
  edge_kernel<<<dim3(NND / 16, NND / 16), 256, 0, stream>>>(buf1, buf3, cvec, we2, be2, S);
  symmetrize_kernel<<<(NND * NND) / 256, 256, 0, stream>>>(S, (float*)d_out);
}